// MINDSCC_44899588112761
// MI455X (gfx1250) — compile-verified
//
#include <hip/hip_runtime.h>
#include <math.h>

// ---------------------------------------------------------------------------
// MIND-SSC descriptor loss for MI455X (gfx1250).
//  k1 (x2): fused diff^2 + separable 5-tap Gaussian (WMMA f32 16x16x4 banded
//           matmul per axis, all in LDS) -> tile-blocked ssd + mind_var partials
//  reduce : deterministic sum of mind_var partials -> global means
//  k3     : fused min/mean/clamp/exp/MSE over both ssd volumes -> loss partials
//  reduce : deterministic final scalar
// ---------------------------------------------------------------------------

typedef __attribute__((ext_vector_type(2))) float v2f;
typedef __attribute__((ext_vector_type(8))) float v8f;

#define VOL    160
#define NVOX   (VOL*VOL*VOL)       // 4,096,000
#define NCH    12
#define NTILES 1000                // 10x10x10 tiles of 16^3
#define SSD_ELEMS 49152000ull      // 12 * 160^3

// 12 stencil pairs: indices into the 6 face-neighbor offsets (SIX - 1)
__device__ __constant__ int c_PA[12] = {1,2,2,3,3,4,4,4,5,5,5,5};
__device__ __constant__ int c_PB[12] = {0,0,1,0,2,1,2,3,0,1,3,4};
__device__ __constant__ int c_SZ[6]  = {-1,0,0,0,1,0};
__device__ __constant__ int c_SY[6]  = {0,0,-1,0,0,1};
__device__ __constant__ int c_SX[6]  = {0,-1,0,1,0,0};

// Normalized Gaussian, sigma=0.8, N=5
#define GW0 0.0219296f
#define GW1 0.2285122f
#define GW2 0.4991161f

__device__ __forceinline__ float band(int t) {
  float r = 0.0f;
  if (t == 2) r = GW2;
  else if (t == 1 || t == 3) r = GW1;
  else if (t == 0 || t == 4) r = GW0;
  return r;
}
__device__ __forceinline__ int clampi(int v, int hi) {
  return v < 0 ? 0 : (v > hi ? hi : v);
}

#if __has_builtin(__builtin_amdgcn_wmma_f32_16x16x4_f32)
#define USE_WMMA 1
#else
#define USE_WMMA 0
#endif

// ---------------------------------------------------------------------------
// Kernel 1: per-volume fused stencil + smoothing. One block = one 16^3 tile.
// ---------------------------------------------------------------------------
__global__ __launch_bounds__(256) void mind_ssd_kernel(
    const float* __restrict__ in,     // [160^3]
    float* __restrict__ ssd_out,      // [12][1000][4096] tile-blocked
    float* __restrict__ part_mv)      // [1000] per-block mind_var partial sums
{
  __shared__ float bufA[8192];  // d[20][20][20] (8000), later sy[20][16][16] (5120)
  __shared__ float bufB[6400];  // sx[20][20][16] (6400), later ssd[16][16][16] (4096)

  const int tid  = threadIdx.x;
  const int wav  = tid >> 5;
  const int lane = tid & 31;
  const int lo   = lane & 15;
  const int hi   = lane >> 4;

  const int t  = blockIdx.x;
  const int tx = (t % 10) * 16;
  const int ty = ((t / 10) % 10) * 16;
  const int tz = (t / 100) * 16;

#if USE_WMMA
  // Banded Gaussian weight matrix A[16][20], A[m][k] = gw[k-m] if 0<=k-m<=4.
  // ISA A-layout (16x4 f32): lanes 0-15 hold M=lane, v0->K0, v1->K1;
  // lanes 16-31 hold M=lane-16, v0->K2, v1->K3.  Same fragment reused for all
  // three axis passes.
  v2f aw[5];
#pragma unroll
  for (int kc = 0; kc < 5; ++kc) {
    int k0 = kc * 4;
    aw[kc][0] = band(k0 + 0 + 2 * hi - lo);
    aw[kc][1] = band(k0 + 1 + 2 * hi - lo);
  }
#endif

  float rmin[16], rsum[16];
#pragma unroll
  for (int i = 0; i < 16; ++i) { rmin[i] = 3.4e38f; rsum[i] = 0.0f; }

  for (int c = 0; c < NCH; ++c) {
    const int ia = c_PA[c], ib = c_PB[c];
    const int az1 = c_SZ[ia], ay1 = c_SY[ia], ax1 = c_SX[ia];
    const int az2 = c_SZ[ib], ay2 = c_SY[ib], ax2 = c_SX[ib];

    // ---- stage 1: d = (shift_a - shift_b)^2 over 20^3 halo region --------
    // Two-level clamp == replicate-pad of d (for smoothing) composed with
    // replicate-pad of the image (for the shifts).  Input fits in L2.
    for (int i = tid; i < 8000; i += 256) {
      int xr = i % 20, yr = (i / 20) % 20, zr = i / 400;
      int zc = clampi(tz + zr - 2, VOL - 1);
      int yc = clampi(ty + yr - 2, VOL - 1);
      int xc = clampi(tx + xr - 2, VOL - 1);
      float va = in[(clampi(zc + az1, VOL-1) * VOL + clampi(yc + ay1, VOL-1)) * VOL + clampi(xc + ax1, VOL-1)];
      float vb = in[(clampi(zc + az2, VOL-1) * VOL + clampi(yc + ay2, VOL-1)) * VOL + clampi(xc + ax2, VOL-1)];
      float df = va - vb;
      bufA[i] = df * df;
    }
    __syncthreads();

#if USE_WMMA
    // ---- stage 2: conv along x: d[20][20][20] -> sx[20][20][16] ----------
    // D[m=xo][n=yline] = sum_k A[xo][k] * B[k=x][n=y].  B mirrors C/D
    // striping: VGPR v holds rows K=v (lanes 0-15) and K=v+2 (lanes 16-31).
    for (int task = wav; task < 40; task += 8) {
      int zz = task >> 1;
      int yb = (task & 1) * 4;          // y-chunks [0,16) and [4,20)
      v8f acc = {};
#pragma unroll
      for (int kc = 0; kc < 5; ++kc) {
        int k0 = kc * 4;
        v2f b;
        b[0] = bufA[zz * 400 + (yb + lo) * 20 + (k0 + 0 + 2 * hi)];
        b[1] = bufA[zz * 400 + (yb + lo) * 20 + (k0 + 1 + 2 * hi)];
        acc = __builtin_amdgcn_wmma_f32_16x16x4_f32(false, aw[kc], false, b,
                                                    (short)0, acc, false, false);
      }
      if (yb == 0 || lo >= 12) {        // avoid duplicate writes in overlap
#pragma unroll
        for (int r = 0; r < 8; ++r)
          bufB[zz * 320 + (yb + lo) * 16 + (r + 8 * hi)] = acc[r];
      }
    }
    __syncthreads();

    // ---- stage 3: conv along y: sx[20][20][16] -> sy[20][16][16] (bufA) --
    for (int task = wav; task < 20; task += 8) {
      int zz = task;
      v8f acc = {};
#pragma unroll
      for (int kc = 0; kc < 5; ++kc) {
        int k0 = kc * 4;
        v2f b;
        b[0] = bufB[zz * 320 + (k0 + 0 + 2 * hi) * 16 + lo];
        b[1] = bufB[zz * 320 + (k0 + 1 + 2 * hi) * 16 + lo];
        acc = __builtin_amdgcn_wmma_f32_16x16x4_f32(false, aw[kc], false, b,
                                                    (short)0, acc, false, false);
      }
#pragma unroll
      for (int r = 0; r < 8; ++r)
        bufA[(zz * 16 + (r + 8 * hi)) * 16 + lo] = acc[r];
    }
    __syncthreads();

    // ---- stage 4: conv along z: sy[20][16][16] -> ssd[16][16][16] (bufB) -
    for (int task = wav; task < 16; task += 8) {
      int yy = task;
      v8f acc = {};
#pragma unroll
      for (int kc = 0; kc < 5; ++kc) {
        int k0 = kc * 4;
        v2f b;
        b[0] = bufA[((k0 + 0 + 2 * hi) * 16 + yy) * 16 + lo];
        b[1] = bufA[((k0 + 1 + 2 * hi) * 16 + yy) * 16 + lo];
        acc = __builtin_amdgcn_wmma_f32_16x16x4_f32(false, aw[kc], false, b,
                                                    (short)0, acc, false, false);
      }
#pragma unroll
      for (int r = 0; r < 8; ++r)
        bufB[((r + 8 * hi) * 16 + yy) * 16 + lo] = acc[r];
    }
    __syncthreads();
#else
    // ---- VALU fallback: separable 5-tap convolutions ---------------------
    for (int i = tid; i < 20 * 20 * 16; i += 256) {       // x-conv
      int xo = i % 16, yr = (i / 16) % 20, zz = i / 320;
      const float* p = &bufA[zz * 400 + yr * 20 + xo];
      bufB[i] = GW0*p[0] + GW1*p[1] + GW2*p[2] + GW1*p[3] + GW0*p[4];
    }
    __syncthreads();
    for (int i = tid; i < 20 * 16 * 16; i += 256) {       // y-conv
      int xo = i % 16, yo = (i / 16) % 16, zz = i / 256;
      const float* p = &bufB[zz * 320 + yo * 16 + xo];
      bufA[i] = GW0*p[0] + GW1*p[16] + GW2*p[32] + GW1*p[48] + GW0*p[64];
    }
    __syncthreads();
    for (int i = tid; i < 16 * 16 * 16; i += 256) {       // z-conv
      const float* p = &bufA[i];
      bufB[i] = GW0*p[0] + GW1*p[256] + GW2*p[512] + GW1*p[768] + GW0*p[1024];
    }
    __syncthreads();
#endif

    // ---- stage 5: running channel min/sum + coalesced tile-blocked store -
    float* gout = ssd_out + ((size_t)(c * NTILES + t)) * 4096;
#pragma unroll
    for (int i = 0; i < 16; ++i) {
      float v = bufB[tid + (i << 8)];
      rmin[i] = fminf(rmin[i], v);
      rsum[i] += v;
      gout[tid + (i << 8)] = v;
    }
    __syncthreads();
  }

  // mind_var partial = sum over owned voxels of (mean_c - min_c)
  float p = 0.0f;
#pragma unroll
  for (int i = 0; i < 16; ++i) p += rsum[i] * (1.0f / 12.0f) - rmin[i];
  bufA[tid] = p;
  __syncthreads();
  for (int s = 128; s > 0; s >>= 1) {
    if (tid < s) bufA[tid] += bufA[tid + s];
    __syncthreads();
  }
  if (tid == 0) part_mv[t] = bufA[0];
}

// ---------------------------------------------------------------------------
// Deterministic reduction of the two mind_var partial arrays -> stats[0..1]
// ---------------------------------------------------------------------------
__global__ __launch_bounds__(256) void reduce_mv_kernel(
    const float* __restrict__ pa, const float* __restrict__ pb,
    float* __restrict__ stats)
{
  __shared__ float ra[256], rb[256];
  int tid = threadIdx.x;
  float sa = 0.0f, sb = 0.0f;
  for (int i = tid; i < NTILES; i += 256) { sa += pa[i]; sb += pb[i]; }
  ra[tid] = sa; rb[tid] = sb;
  __syncthreads();
  for (int s = 128; s > 0; s >>= 1) {
    if (tid < s) { ra[tid] += ra[tid + s]; rb[tid] += rb[tid + s]; }
    __syncthreads();
  }
  if (tid == 0) { stats[0] = ra[0]; stats[1] = rb[0]; }
}

// ---------------------------------------------------------------------------
// Kernel 3: fused min/mean/clamp/exp + squared-diff partials per tile.
// ---------------------------------------------------------------------------
__global__ __launch_bounds__(256) void mind_loss_kernel(
    const float* __restrict__ ssd_x, const float* __restrict__ ssd_y,
    const float* __restrict__ stats, float* __restrict__ part_loss)
{
  __shared__ float red[256];
  const int tid = threadIdx.x;
  const int t   = blockIdx.x;
  const float mx = stats[0] * (1.0f / (float)NVOX);
  const float my = stats[1] * (1.0f / (float)NVOX);
  const float mx_lo = mx * 0.001f, mx_hi = mx * 1000.0f;
  const float my_lo = my * 0.001f, my_hi = my * 1000.0f;

  float acc = 0.0f;
  for (int i = 0; i < 16; ++i) {
    int v = tid + (i << 8);
    float sxv[12], syv[12];
    float minx = 3.4e38f, sumx = 0.0f, miny = 3.4e38f, sumy = 0.0f;
#pragma unroll
    for (int c = 0; c < 12; ++c) {
      float a = ssd_x[((size_t)(c * NTILES + t)) * 4096 + v];
      float b = ssd_y[((size_t)(c * NTILES + t)) * 4096 + v];
      sxv[c] = a; syv[c] = b;
      minx = fminf(minx, a); sumx += a;
      miny = fminf(miny, b); sumy += b;
    }
    float mvx = fminf(fmaxf(sumx * (1.0f / 12.0f) - minx, mx_lo), mx_hi);
    float mvy = fminf(fmaxf(sumy * (1.0f / 12.0f) - miny, my_lo), my_hi);
    float rix = -1.0f / mvx, riy = -1.0f / mvy;
#pragma unroll
    for (int c = 0; c < 12; ++c) {
      float dx = __expf((sxv[c] - minx) * rix);
      float dy = __expf((syv[c] - miny) * riy);
      float d = dx - dy;
      acc += d * d;
    }
  }
  red[tid] = acc;
  __syncthreads();
  for (int s = 128; s > 0; s >>= 1) {
    if (tid < s) red[tid] += red[tid + s];
    __syncthreads();
  }
  if (tid == 0) part_loss[t] = red[0];
}

// ---------------------------------------------------------------------------
// Deterministic final reduction -> scalar loss.
// ---------------------------------------------------------------------------
__global__ __launch_bounds__(256) void reduce_loss_kernel(
    const float* __restrict__ pl, float* __restrict__ out)
{
  __shared__ float r[256];
  int tid = threadIdx.x;
  float s0 = 0.0f;
  for (int i = tid; i < NTILES; i += 256) s0 += pl[i];
  r[tid] = s0;
  __syncthreads();
  for (int s = 128; s > 0; s >>= 1) {
    if (tid < s) r[tid] += r[tid + s];
    __syncthreads();
  }
  if (tid == 0) out[0] = r[0] * (1.0f / 49152000.0f);
}

// ---------------------------------------------------------------------------
extern "C" void kernel_launch(void* const* d_in, const int* in_sizes, int n_in,
                              void* d_out, int out_size, void* d_ws, size_t ws_size,
                              hipStream_t stream) {
  (void)in_sizes; (void)n_in; (void)out_size; (void)ws_size;
  const float* x = (const float*)d_in[0];
  const float* y = (const float*)d_in[1];
  float* out = (float*)d_out;

  float* ws        = (float*)d_ws;
  float* part_mv_x = ws;                 // 1024
  float* part_mv_y = ws + 1024;          // 1024
  float* part_loss = ws + 2048;          // 1024
  float* stats     = ws + 3072;          // 8
  float* ssd_x     = ws + 4096;          // 49,152,000 floats
  float* ssd_y     = ssd_x + SSD_ELEMS;  // 49,152,000 floats

  dim3 blk(256);
  mind_ssd_kernel<<<NTILES, blk, 0, stream>>>(x, ssd_x, part_mv_x);
  mind_ssd_kernel<<<NTILES, blk, 0, stream>>>(y, ssd_y, part_mv_y);
  reduce_mv_kernel<<<1, blk, 0, stream>>>(part_mv_x, part_mv_y, stats);
  mind_loss_kernel<<<NTILES, blk, 0, stream>>>(ssd_x, ssd_y, stats, part_loss);
  reduce_loss_kernel<<<1, blk, 0, stream>>>(part_loss, out);
}